// DetectionLayer_53747220742217
// MI455X (gfx1250) — compile-verified
//
#include <hip/hip_runtime.h>
#include <cstdint>
#include <cstddef>

namespace {

constexpr int   kN       = 1000;  // ROIs per image
constexpr int   kC       = 81;    // classes
constexpr int   kMaxInst = 100;
constexpr float kMinConf = 0.7f;
constexpr float kNmsThr  = 0.3f;
constexpr int   kThreads = 1024;  // 32 wave32 waves per WGP

__global__ __launch_bounds__(kThreads)
void detection_layer_kernel(const float* __restrict__ rois,    // (B,N,4)
                            const float* __restrict__ probs,   // (B,N,C)
                            const float* __restrict__ deltas,  // (B,N,C,4)
                            const float* __restrict__ meta,    // (B,meta_w)
                            int meta_w,
                            float* __restrict__ out)           // (B,100,6)
{
  const int b   = blockIdx.x;
  const int tid = threadIdx.x;
  const bool active = (tid < kN);

  // LDS-resident per-image state (~52 KB of the 320 KB WGP LDS)
  __shared__ float s_box[kN * 4];   // raw rois -> refined boxes (unsorted)
  __shared__ float s_score[kN];
  __shared__ int   s_cls[kN];
  __shared__ float t_box[kN * 4];   // sorted by descending score
  __shared__ float t_score[kN];
  __shared__ int   t_cls[kN];
  __shared__ int   s_keep[kN];

  // ---- Phase 0: async DMA the ROI tile into LDS (CDNA5 ASYNCcnt path) ----
  if (active) {
    unsigned lds_dst = (unsigned)(uintptr_t)(&s_box[tid * 4]);
    uint64_t gaddr   = (uint64_t)(uintptr_t)(rois + ((size_t)b * kN + tid) * 4);
    asm volatile("global_load_async_to_lds_b128 %0, %1, off"
                 :: "v"(lds_dst), "v"(gaddr)
                 : "memory");
  }

  // ---- window from image_meta (reference: shape from row 0, window per-b) ----
  const float* mrow = meta + (size_t)b * meta_w;
  const float ih = meta[4];
  const float iw = meta[5];
  const float sy = ih - 1.0f, sx = iw - 1.0f;
  const float wy1 = (mrow[7]        ) / sy;
  const float wx1 = (mrow[8]        ) / sx;
  const float wy2 = (mrow[9]  - 1.0f) / sy;
  const float wx2 = (mrow[10] - 1.0f) / sx;

  // ---- Phase 1: per-ROI argmax over classes + gather class-specific deltas ----
  float best = -1.0f;
  int   bid  = 0;
  float dy = 0.0f, dx = 0.0f, dh = 0.0f, dw = 0.0f;
  if (active) {
    const float* prow = probs + ((size_t)b * kN + tid) * kC;
    __builtin_prefetch(prow, 0, 1);                 // global_prefetch_b8
    best = prow[0];
    for (int c = 1; c < kC; ++c) {
      float p = prow[c];
      if (p > best) { best = p; bid = c; }          // strict > == jnp.argmax (first max)
    }
    const float* drow = deltas + (((size_t)b * kN + tid) * kC + bid) * 4;
    dy = drow[0] * 0.1f;
    dx = drow[1] * 0.1f;
    dh = drow[2] * 0.2f;
    dw = drow[3] * 0.2f;
  }

  // drain the async ROI copy, make LDS visible block-wide
  asm volatile("s_wait_asynccnt 0x0" ::: "memory");
  __syncthreads();

  // ---- Phase 2: apply deltas, clip to window, validity mask ----
  if (active) {
    float y1 = s_box[tid * 4 + 0];
    float x1 = s_box[tid * 4 + 1];
    float y2 = s_box[tid * 4 + 2];
    float x2 = s_box[tid * 4 + 3];
    float hh = y2 - y1, ww = x2 - x1;
    float cy = y1 + 0.5f * hh + dy * hh;
    float cx = x1 + 0.5f * ww + dx * ww;
    hh *= expf(dh);
    ww *= expf(dw);
    y1 = cy - 0.5f * hh; x1 = cx - 0.5f * ww;
    y2 = cy + 0.5f * hh; x2 = cx + 0.5f * ww;
    y1 = fminf(fmaxf(y1, wy1), wy2);
    x1 = fminf(fmaxf(x1, wx1), wx2);
    y2 = fminf(fmaxf(y2, wy1), wy2);
    x2 = fminf(fmaxf(x2, wx1), wx2);
    s_box[tid * 4 + 0] = y1;
    s_box[tid * 4 + 1] = x1;
    s_box[tid * 4 + 2] = y2;
    s_box[tid * 4 + 3] = x2;
    const bool valid = (bid > 0) && (best >= kMinConf);
    s_score[tid] = valid ? best : -1.0f;
    s_cls[tid]   = bid;
  }
  __syncthreads();

  // ---- Phase 3: stable descending rank-sort (matches argsort(-scores)) ----
  if (active) {
    const float si = s_score[tid];
    int r = 0;
    for (int j = 0; j < kN; ++j) {
      float sj = s_score[j];
      r += (sj > si) || (sj == si && j < tid);
    }
    t_score[r]       = si;
    t_cls[r]         = s_cls[tid];
    t_box[r * 4 + 0] = s_box[tid * 4 + 0];
    t_box[r * 4 + 1] = s_box[tid * 4 + 1];
    t_box[r * 4 + 2] = s_box[tid * 4 + 2];
    t_box[r * 4 + 3] = s_box[tid * 4 + 3];
    s_keep[r]        = (si > -0.5f) ? 1 : 0;
  }

  // ---- Phase 4: greedy class-aware NMS (sequential over pivots) ----
  for (int i = 0; i < kN - 1; ++i) {
    __syncthreads();                       // publish previous iteration's kills
    if (!s_keep[i]) continue;              // uniform across block
    const float iy1 = t_box[i * 4 + 0];
    const float ix1 = t_box[i * 4 + 1];
    const float iy2 = t_box[i * 4 + 2];
    const float ix2 = t_box[i * 4 + 3];
    const float ai  = (iy2 - iy1) * (ix2 - ix1);
    const int   ci  = t_cls[i];
    for (int j = i + 1 + tid; j < kN; j += kThreads) {
      if (!s_keep[j] || t_cls[j] != ci) continue;
      const float jy1 = t_box[j * 4 + 0];
      const float jx1 = t_box[j * 4 + 1];
      const float jy2 = t_box[j * 4 + 2];
      const float jx2 = t_box[j * 4 + 3];
      const float aj  = (jy2 - jy1) * (jx2 - jx1);
      const float yy1 = fmaxf(iy1, jy1);
      const float xx1 = fmaxf(ix1, jx1);
      const float yy2 = fminf(iy2, jy2);
      const float xx2 = fminf(ix2, jx2);
      const float inter = fmaxf(yy2 - yy1, 0.0f) * fmaxf(xx2 - xx1, 0.0f);
      const float uni   = fmaxf(ai + aj - inter, 1e-10f);
      if (inter / uni > kNmsThr) s_keep[j] = 0;
    }
  }
  __syncthreads();

  // ---- Phase 5: top-100 == in-order compaction of survivors; zero-pad ----
  float* orow = out + (size_t)b * kMaxInst * 6;
  for (int k = tid; k < kMaxInst * 6; k += kThreads) orow[k] = 0.0f;
  __syncthreads();
  if (active && s_keep[tid]) {
    int pos = 0;
    for (int j = 0; j < tid; ++j) pos += s_keep[j];
    if (pos < kMaxInst) {
      orow[pos * 6 + 0] = t_box[tid * 4 + 0];
      orow[pos * 6 + 1] = t_box[tid * 4 + 1];
      orow[pos * 6 + 2] = t_box[tid * 4 + 2];
      orow[pos * 6 + 3] = t_box[tid * 4 + 3];
      orow[pos * 6 + 4] = (float)t_cls[tid];
      orow[pos * 6 + 5] = t_score[tid];
    }
  }
}

} // anonymous namespace

extern "C" void kernel_launch(void* const* d_in, const int* in_sizes, int n_in,
                              void* d_out, int out_size, void* d_ws, size_t ws_size,
                              hipStream_t stream) {
  (void)n_in; (void)out_size; (void)d_ws; (void)ws_size;
  const float* rois   = (const float*)d_in[0];  // (B,N,4)
  const float* probs  = (const float*)d_in[1];  // (B,N,81)
  const float* deltas = (const float*)d_in[2];  // (B,N,81,4)
  const float* meta   = (const float*)d_in[3];  // (B,meta_w)
  const int B      = in_sizes[0] / (kN * 4);
  const int meta_w = in_sizes[3] / B;
  detection_layer_kernel<<<B, kThreads, 0, stream>>>(rois, probs, deltas, meta,
                                                     meta_w, (float*)d_out);
}